// SpatialEncoding_18691697672325
// MI455X (gfx1250) — compile-verified
//
#include <hip/hip_runtime.h>
#include <stdint.h>

// SpatialEncoding: out[b] = broadcast(weight[511]); out[b, n1, n2] = weight[pdm[b,e]]
// B=64, T=1024, E=200000, MAX_PATH_DISTANCE=512. Input x is unused by the math.

#define BATCHES 64
#define TNODES  1024
#define EDGES   200000
#define MPD     512

typedef float v4f __attribute__((ext_vector_type(4)));
typedef int   v4i __attribute__((ext_vector_type(4)));

__global__ __launch_bounds__(1024)
void spatial_encoding_fused(const float* __restrict__ weight,
                            const int*   __restrict__ pdm,
                            const int*   __restrict__ eim,
                            float*       __restrict__ out)
{
    __shared__ float w_lds[MPD];

    const int b   = blockIdx.x;    // one workgroup owns one batch slice
    const int tid = threadIdx.x;   // 0..1023 (32 wave32s)

    // ---- Stage the 2KB embedding table into LDS via the CDNA5 async path.
    // Warps 0..15 (tid<512) each move one f32 from global weight[tid] into w_lds[tid] in LDS.
    if (tid < MPD) {
        uint32_t lds_off = (uint32_t)(uintptr_t)&w_lds[tid]; // addrspace(3) byte offset
        uint32_t goff    = (uint32_t)tid * 4u;               // per-lane byte offset
        asm volatile("global_load_async_to_lds_b32 %0, %1, %2"
                     :: "v"(lds_off), "v"(goff),
                        "s"((uint64_t)(uintptr_t)weight)
                     : "memory");
        asm volatile("s_wait_asynccnt 0" ::: "memory");
    }
    __syncthreads();

    const float fv    = w_lds[MPD - 1];          // 'no path' encoding
    float*      slice = out + (size_t)b * TNODES * TNODES;

    // ---- Phase 1: fill the 4MB slice with b128 stores.
    // T*T/4 = 262144 float4; exactly 256 per thread.
    v4f  fill4  = {fv, fv, fv, fv};
    v4f* slice4 = (v4f*)slice;
    #pragma unroll 4
    for (int i = tid; i < (TNODES * TNODES) / 4; i += 1024)
        slice4[i] = fill4;

    // ---- Order fill stores before scatter stores (this block owns the slice).
    asm volatile("s_wait_storecnt 0" ::: "memory");
    __threadfence();
    __syncthreads();

    // ---- Phase 2: scatter, 4 edges per thread per iteration.
    // Index streams via b128 non-temporal loads (don't pollute L2 — we want it
    // holding the output slices); embedding gather from LDS; 4B scatter stores
    // into the still-L2-resident slice.
    const v4i* pdm4 = (const v4i*)(pdm + (size_t)b * EDGES);      // 16B aligned
    const v4i* eim4 = (const v4i*)(eim + (size_t)b * 2 * EDGES);  // 16B aligned

    const int E4 = EDGES / 4;  // 50000, exact
    for (int i = tid; i < E4; i += 1024) {
        v4i d4 = __builtin_nontemporal_load(&pdm4[i]);        // 4 distances
        v4i p0 = __builtin_nontemporal_load(&eim4[2 * i]);    // (n1,n2) x2
        v4i p1 = __builtin_nontemporal_load(&eim4[2 * i + 1]);// (n1,n2) x2

        slice[p0.x * TNODES + p0.y] = w_lds[d4.x & (MPD - 1)];
        slice[p0.z * TNODES + p0.w] = w_lds[d4.y & (MPD - 1)];
        slice[p1.x * TNODES + p1.y] = w_lds[d4.z & (MPD - 1)];
        slice[p1.z * TNODES + p1.w] = w_lds[d4.w & (MPD - 1)];
    }
}

extern "C" void kernel_launch(void* const* d_in, const int* in_sizes, int n_in,
                              void* d_out, int out_size, void* d_ws, size_t ws_size,
                              hipStream_t stream) {
    // setup_inputs order: x (unused), weight, path_distance_map, edge_index_map
    const float* weight = (const float*)d_in[1];
    const int*   pdm    = (const int*)d_in[2];
    const int*   eim    = (const int*)d_in[3];
    float*       out    = (float*)d_out;

    spatial_encoding_fused<<<dim3(BATCHES), dim3(1024), 0, stream>>>(weight, pdm, eim, out);
}